// TemporalGNN_30365418783390
// MI455X (gfx1250) — compile-verified
//
#include <hip/hip_runtime.h>

#define N_NODES 50000    // multiple of 16 -> GEMM tiles are always full
#define N_EDGES 800000
#define HID     128
#define BN_EPS  1e-5f

typedef float v2f __attribute__((ext_vector_type(2)));
typedef float v8f __attribute__((ext_vector_type(8)));

static_assert(N_NODES % 16 == 0, "tile path assumes full 16-row stripes");
static_assert(HID == 128, "layouts assume 128 channels");

// ---------------------------------------------------------------------------
// Degree / normalization
// ---------------------------------------------------------------------------
__global__ void k_deg_init(float* __restrict__ deg, int n) {
    int i = blockIdx.x * blockDim.x + threadIdx.x;
    if (i < n) deg[i] = 1.0f;                       // self-loop contributes 1
}

__global__ void k_deg_edges(const int* __restrict__ col, float* __restrict__ deg, int ne) {
    int e = blockIdx.x * blockDim.x + threadIdx.x;
    if (e < ne) atomicAdd(&deg[col[e]], 1.0f);
}

__global__ void k_deg_rsqrt(float* __restrict__ deg, int n) {
    int i = blockIdx.x * blockDim.x + threadIdx.x;
    if (i < n) deg[i] = rsqrtf(deg[i]);             // deg >= 1 always
}

// ---------------------------------------------------------------------------
// Dense transform: Out[n,128] = A[n,128] @ W[128,128] via V_WMMA_F32_16X16X4_F32
// One block = 256 threads = 8 waves; block handles a full 16-row stripe, wave w
// produces the 16x16 tile at columns [16w, 16w+16). A stripe staged in LDS.
// No tail handling: N is an exact multiple of 16.
// ---------------------------------------------------------------------------
__global__ __launch_bounds__(256) void k_gemm_wmma(const float* __restrict__ A,
                                                   const float* __restrict__ W,
                                                   float* __restrict__ Out) {
    __shared__ float As[16 * HID];                  // 8 KB per block
    const int m0  = blockIdx.x * 16;
    const int tid = threadIdx.x;

    // Cooperative, coalesced load of the 16x128 A stripe (2048 floats).
    {
        const float4* src = (const float4*)(A + (size_t)m0 * HID);
        float4*       dst = (float4*)As;
        dst[tid]       = src[tid];
        dst[tid + 256] = src[tid + 256];
    }
    __syncthreads();

    const int wave  = tid >> 5;
    const int lane  = tid & 31;
    const int n0    = wave * 16;
    const int lhalf = lane >> 4;                    // 0: lanes 0-15, 1: lanes 16-31
    const int l16   = lane & 15;

    v8f c = {0.f, 0.f, 0.f, 0.f, 0.f, 0.f, 0.f, 0.f};

#pragma unroll 8
    for (int k = 0; k < HID; k += 4) {
        // A frag (16x4 f32): lanes 0-15 carry K=k,k+1; lanes 16-31 carry K=k+2,k+3
        const int ak = k + lhalf * 2;
        v2f a;
        a.x = As[l16 * HID + ak];
        a.y = As[l16 * HID + ak + 1];
        // B frag (4x16 f32): VGPR0 -> rows K=k / K=k+2, VGPR1 -> rows K=k+1 / K=k+3
        v2f b;
        b.x = W[(ak)     * HID + n0 + l16];
        b.y = W[(ak + 1) * HID + n0 + l16];
        c = __builtin_amdgcn_wmma_f32_16x16x4_f32(false, a, false, b,
                                                  (short)0, c, false, false);
    }

    // C/D layout: VGPR v holds row m0 + v + 8*lhalf, column n0 + l16.
    // Unconditional: every tile is full. Each store = two 64B coalesced runs.
    float* outb = Out + (size_t)(m0 + 8 * lhalf) * HID + n0 + l16;
#pragma unroll
    for (int v = 0; v < 8; ++v) {
        outb[(size_t)v * HID] = c[v];
    }
}

// ---------------------------------------------------------------------------
// agg[i,:] = xw[i,:] * dinv[i]^2   (self-loop term; also initializes agg)
// ---------------------------------------------------------------------------
__global__ void k_self_init(const float* __restrict__ xw,
                            const float* __restrict__ dinv,
                            float* __restrict__ agg, int n) {
    int t = blockIdx.x * blockDim.x + threadIdx.x;      // over n*128/4 float4s
    int total = n * (HID / 4);
    if (t >= total) return;
    int node = t / (HID / 4);
    float s = dinv[node];
    s = s * s;
    float4 v = ((const float4*)xw)[t];
    v.x *= s; v.y *= s; v.z *= s; v.w *= s;
    ((float4*)agg)[t] = v;
}

// ---------------------------------------------------------------------------
// Edge scatter: one wave per edge; lane handles 4 channels (float4 read,
// 4x global_atomic_add_f32). Feature buffers are L2-resident (~77MB < 192MB).
// ---------------------------------------------------------------------------
__global__ __launch_bounds__(256) void k_edge_scatter(const int* __restrict__ row,
                                                      const int* __restrict__ col,
                                                      const float* __restrict__ dinv,
                                                      const float* __restrict__ xw,
                                                      float* __restrict__ agg, int ne) {
    int e    = (blockIdx.x * blockDim.x + threadIdx.x) >> 5;
    int lane = threadIdx.x & 31;
    if (e >= ne) return;
    int r = row[e], c = col[e];
    float nrm = dinv[r] * dinv[c];
    float4 v = ((const float4*)(xw + (size_t)r * HID))[lane];
    float* d = agg + (size_t)c * HID + lane * 4;
    atomicAdd(d + 0, v.x * nrm);
    atomicAdd(d + 1, v.y * nrm);
    atomicAdd(d + 2, v.z * nrm);
    atomicAdd(d + 3, v.w * nrm);
}

// ---------------------------------------------------------------------------
// Epilogues: BN(bias folded)+ReLU, and plain bias+ReLU for layer 3.
// y = (agg + b - m) * g * rsqrt(v+eps) + beta  ==  agg*s + t
// ---------------------------------------------------------------------------
__global__ void k_bn_relu(const float* __restrict__ agg,
                          const float* __restrict__ b, const float* __restrict__ g,
                          const float* __restrict__ beta, const float* __restrict__ m,
                          const float* __restrict__ v, float* __restrict__ out, int n) {
    int i = blockIdx.x * blockDim.x + threadIdx.x;      // over n*128
    if (i >= n * HID) return;
    int ch = i & (HID - 1);
    float s = g[ch] * rsqrtf(v[ch] + BN_EPS);
    float t = (b[ch] - m[ch]) * s + beta[ch];
    float y = fmaf(agg[i], s, t);
    out[i] = fmaxf(y, 0.0f);
}

__global__ void k_bias_relu(const float* __restrict__ agg,
                            const float* __restrict__ b,
                            float* __restrict__ out, int n) {
    int i = blockIdx.x * blockDim.x + threadIdx.x;
    if (i >= n * HID) return;
    out[i] = fmaxf(agg[i] + b[i & (HID - 1)], 0.0f);
}

// ---------------------------------------------------------------------------
// Head: out[n,2] = h @ fcW + fcb. One wave per node, shuffle reduction.
// ---------------------------------------------------------------------------
__global__ __launch_bounds__(128) void k_fc_head(const float* __restrict__ h,
                                                 const float* __restrict__ fcW,
                                                 const float* __restrict__ fcb,
                                                 float* __restrict__ out, int n) {
    int node = (blockIdx.x * blockDim.x + threadIdx.x) >> 5;
    int lane = threadIdx.x & 31;
    if (node >= n) return;
    float4 hv = ((const float4*)(h + (size_t)node * HID))[lane];
    float s0 = 0.f, s1 = 0.f;
    int kb = lane * 4;
#pragma unroll
    for (int j = 0; j < 4; ++j) {
        float hx = (j == 0) ? hv.x : (j == 1) ? hv.y : (j == 2) ? hv.z : hv.w;
        s0 = fmaf(hx, fcW[(kb + j) * 2 + 0], s0);
        s1 = fmaf(hx, fcW[(kb + j) * 2 + 1], s1);
    }
#pragma unroll
    for (int off = 16; off > 0; off >>= 1) {
        s0 += __shfl_down(s0, off, 32);
        s1 += __shfl_down(s1, off, 32);
    }
    if (lane == 0) {
        out[(size_t)node * 2 + 0] = s0 + fcb[0];
        out[(size_t)node * 2 + 1] = s1 + fcb[1];
    }
}

// ---------------------------------------------------------------------------
extern "C" void kernel_launch(void* const* d_in, const int* in_sizes, int n_in,
                              void* d_out, int out_size, void* d_ws, size_t ws_size,
                              hipStream_t stream) {
    const float* x    = (const float*)d_in[0];
    const int*   ei   = (const int*)d_in[1];
    const int*   row  = ei;
    const int*   col  = ei + N_EDGES;
    const float* W1   = (const float*)d_in[2];
    const float* b1   = (const float*)d_in[3];
    const float* g1   = (const float*)d_in[4];
    const float* be1  = (const float*)d_in[5];
    const float* m1   = (const float*)d_in[6];
    const float* v1   = (const float*)d_in[7];
    const float* W2   = (const float*)d_in[8];
    const float* b2   = (const float*)d_in[9];
    const float* g2   = (const float*)d_in[10];
    const float* be2  = (const float*)d_in[11];
    const float* m2   = (const float*)d_in[12];
    const float* v2   = (const float*)d_in[13];
    const float* W3   = (const float*)d_in[14];
    const float* b3   = (const float*)d_in[15];
    const float* fcW  = (const float*)d_in[16];
    const float* fcb  = (const float*)d_in[17];
    float*       out  = (float*)d_out;

    // Workspace layout: dinv[N] | T0[N*128] | T1[N*128] | T2[N*128]
    const size_t feat = (size_t)N_NODES * HID;
    float* dinv = (float*)d_ws;
    float* T0 = (float*)((char*)d_ws + (((size_t)N_NODES * 4 + 511) & ~(size_t)511));
    float* T1 = T0 + feat;
    float* T2 = T1 + feat;

    const int TPB = 256;
    const int nBlkN    = (N_NODES + TPB - 1) / TPB;
    const int nBlkE    = (N_EDGES + TPB - 1) / TPB;
    const int nBlkFeat = (int)((feat + TPB - 1) / TPB);
    const int nBlkF4   = (int)((feat / 4 + TPB - 1) / TPB);
    const int nBlkTile = N_NODES / 16;                     // 3125 full tiles
    const int nBlkScat = (int)(((size_t)N_EDGES * 32 + TPB - 1) / TPB);
    const int nBlkHead = (int)(((size_t)N_NODES * 32 + 127) / 128);

    // --- normalization (shared by all 3 layers) ---
    k_deg_init  <<<nBlkN, TPB, 0, stream>>>(dinv, N_NODES);
    k_deg_edges <<<nBlkE, TPB, 0, stream>>>(col, dinv, N_EDGES);
    k_deg_rsqrt <<<nBlkN, TPB, 0, stream>>>(dinv, N_NODES);

    // --- layer 1 ---
    k_gemm_wmma    <<<nBlkTile, TPB, 0, stream>>>(x, W1, T0);
    k_self_init    <<<nBlkF4,   TPB, 0, stream>>>(T0, dinv, T1, N_NODES);
    k_edge_scatter <<<nBlkScat, TPB, 0, stream>>>(row, col, dinv, T0, T1, N_EDGES);
    k_bn_relu      <<<nBlkFeat, TPB, 0, stream>>>(T1, b1, g1, be1, m1, v1, T2, N_NODES);

    // --- layer 2 ---
    k_gemm_wmma    <<<nBlkTile, TPB, 0, stream>>>(T2, W2, T0);
    k_self_init    <<<nBlkF4,   TPB, 0, stream>>>(T0, dinv, T1, N_NODES);
    k_edge_scatter <<<nBlkScat, TPB, 0, stream>>>(row, col, dinv, T0, T1, N_EDGES);
    k_bn_relu      <<<nBlkFeat, TPB, 0, stream>>>(T1, b2, g2, be2, m2, v2, T2, N_NODES);

    // --- layer 3 (no BN) ---
    k_gemm_wmma    <<<nBlkTile, TPB, 0, stream>>>(T2, W3, T0);
    k_self_init    <<<nBlkF4,   TPB, 0, stream>>>(T0, dinv, T1, N_NODES);
    k_edge_scatter <<<nBlkScat, TPB, 0, stream>>>(row, col, dinv, T0, T1, N_EDGES);
    k_bias_relu    <<<nBlkFeat, TPB, 0, stream>>>(T1, b3, T2, N_NODES);

    // --- head ---
    k_fc_head <<<nBlkHead, 128, 0, stream>>>(T2, fcW, fcb, out, N_NODES);
}